// EdgeAwareGAT_22832046145626
// MI455X (gfx1250) — compile-verified
//
#include <hip/hip_runtime.h>
#include <hip/hip_bf16.h>
#include <cstdint>

// ---------------------------------------------------------------------------
// EdgeAwareGAT for MI455X (gfx1250): fp32 WMMA (V_WMMA_F32_16X16X4_F32) for
// the dense feature GEMMs; L2-resident atomic scatter/gather for attention.
// ---------------------------------------------------------------------------

typedef __attribute__((ext_vector_type(2))) float v2f;
typedef __attribute__((ext_vector_type(8))) float v8f;

constexpr int NNODE = 50000;
constexpr int NEDGE = 800000;
constexpr int HEADS = 8;
constexpr int CIN   = 128;
constexpr int CHID  = 64;             // per-head hidden
constexpr int HC    = HEADS * CHID;   // 512
constexpr int COUT  = 128;
constexpr int EDIM  = 4;
constexpr int NGRAPH = 64;

// ---------------------------------------------------------------------------
// fp32 WMMA GEMM:  C[N x M] = A[N x K] @ B[K x M]
// grid = (M/128, ceil(N/64)), block = (32, 2).
// Each wave computes a 64x64 strip: 4 row-tiles x 4 col-tiles = 16 v8f accs
// (128 VGPRs of accumulators). __launch_bounds__(64,1) lifts the default
// VGPR/occupancy cap so all accumulators stay resident (no scratch spills);
// 16 independent WMMA chains per wave already hide XDL + load latency.
// Per k-step: 4 A b64 loads + 8 B b32 loads feed 16 WMMAs (0.75 loads/WMMA).
// Requires K % 4 == 0, M % 128 == 0, N % 16 == 0 (N=50000 -> 3125 tiles).
// Last row-block: loads clamp to row N-1; stores guarded per 16-row tile with
// a wave-uniform condition so EXEC stays all-ones around the WMMAs.
// A-layout (16x4 f32): lane<16 holds row=lane, k={0,1}; lane>=16 k={2,3}.
// B-layout (4x16 f32): lane<16 holds col=lane, k={0,1}; lane>=16 k={2,3}.
// D-layout (16x16 f32): vgpr r -> row r (lanes 0-15) / row r+8 (lanes 16-31).
// ---------------------------------------------------------------------------
__global__ void __launch_bounds__(64, 1)
k_gemm_f32wmma(const float* __restrict__ A,
               const float* __restrict__ B,
               float* __restrict__ C,
               int Nr, int K, int M) {
  const int lane = threadIdx.x;            // 0..31
  const int wid  = threadIdx.y;            // 0..1
  const int r    = lane & 15;
  const int hi   = lane >> 4;              // 0/1
  const int kb   = hi * 2;                 // k sub-offset for this half-wave
  const int rowblk  = blockIdx.y * 64;
  const int colbase = (blockIdx.x * 2 + wid) * 64;

  const float* Ap[4];
#pragma unroll
  for (int rt = 0; rt < 4; ++rt) {
    int row = rowblk + rt * 16 + r;
    if (row >= Nr) row = Nr - 1;           // clamped dummy read for tail block
    Ap[rt] = A + (size_t)row * K;
  }

  v8f acc[16] = {};

  for (int k0 = 0; k0 < K; k0 += 4) {
    v2f a[4];
#pragma unroll
    for (int rt = 0; rt < 4; ++rt) {
      a[rt].x = Ap[rt][k0 + kb];
      a[rt].y = Ap[rt][k0 + kb + 1];
    }
    const float* B0 = B + (size_t)(k0 + kb) * M + colbase + r;
    const float* B1 = B0 + M;
#pragma unroll
    for (int t = 0; t < 4; ++t) {
      v2f b;
      b.x = B0[t * 16];
      b.y = B1[t * 16];
#pragma unroll
      for (int rt = 0; rt < 4; ++rt)
        acc[rt * 4 + t] = __builtin_amdgcn_wmma_f32_16x16x4_f32(
            false, a[rt], false, b, (short)0, acc[rt * 4 + t], false, false);
    }
  }

  const int hi8 = hi * 8;
#pragma unroll
  for (int rt = 0; rt < 4; ++rt) {
    const int tbase = rowblk + rt * 16;
    if (tbase < Nr) {                      // uniform: tile fully in range (N%16==0)
      const size_t m0 = (size_t)tbase + hi8;
#pragma unroll
      for (int t = 0; t < 4; ++t) {
        const int col = colbase + t * 16 + r;
#pragma unroll
        for (int rr = 0; rr < 8; ++rr)
          C[(m0 + rr) * M + col] = acc[rt * 4 + t][rr];
      }
    }
  }
}

// ---------------------------------------------------------------------------
// Float atomic max via sign-split int/uint trick (init with -1e30).
// ---------------------------------------------------------------------------
__device__ __forceinline__ void atomicMaxF(float* addr, float v) {
  if (v >= 0.0f) atomicMax((int*)addr, __float_as_int(v));
  else           atomicMin((unsigned int*)addr, __float_as_uint(v));
}

__global__ void k_fill(float* __restrict__ p, float v, int n) {
  int i = blockIdx.x * blockDim.x + threadIdx.x;
  if (i < n) p[i] = v;
}

// self-loop edge attr: mean of incoming edge attrs per dst node
__global__ void k_loop_acc(const int* __restrict__ ei, const float* __restrict__ eattr,
                           float* __restrict__ loop_sum, float* __restrict__ deg) {
  int e = blockIdx.x * blockDim.x + threadIdx.x;
  if (e >= NEDGE) return;
  int dN = ei[NEDGE + e];
  atomicAdd(&deg[dN], 1.0f);
#pragma unroll
  for (int d = 0; d < EDIM; ++d)
    atomicAdd(&loop_sum[(size_t)dN * EDIM + d], eattr[(size_t)e * EDIM + d]);
}

__global__ void k_loop_div(float* __restrict__ loop_sum, const float* __restrict__ deg) {
  int i = blockIdx.x * blockDim.x + threadIdx.x;
  if (i >= NNODE * EDIM) return;
  float c = deg[i / EDIM];
  loop_sum[i] /= (c > 1.0f ? c : 1.0f);
}

// q[d][h] = sum_c We[d][h*ch+c] * a_e[h][c]   (collapses the edge-feature GEMM)
__global__ void k_compute_q(const float* __restrict__ We, const float* __restrict__ ae,
                            float* __restrict__ q, int heads, int ch) {
  int i = threadIdx.x;
  if (i >= EDIM * heads) return;
  int d = i / heads, hd = i - d * heads;
  float s = 0.0f;
  for (int c = 0; c < ch; ++c)
    s += We[(size_t)d * heads * ch + hd * ch + c] * ae[hd * ch + c];
  q[d * heads + hd] = s;
}

// per-node attention dots: as_n[n][h] = h[n,h,:].a_s[h], ad_n likewise
__global__ void k_node_alpha(const float* __restrict__ feat,
                             const float* __restrict__ a_s, const float* __restrict__ a_d,
                             float* __restrict__ as_n, float* __restrict__ ad_n,
                             int heads, int ch) {
  int i = blockIdx.x * blockDim.x + threadIdx.x;
  if (i >= NNODE * heads) return;
  int n = i / heads, hd = i - n * heads;
  const float* hp = feat + (size_t)n * heads * ch + (size_t)hd * ch;
  float s = 0.0f, d = 0.0f;
  for (int c = 0; c < ch; ++c) {
    float v = hp[c];
    s += v * a_s[hd * ch + c];
    d += v * a_d[hd * ch + c];
  }
  as_n[i] = s;
  ad_n[i] = d;
}

// per-(edge,head) raw attention + leaky_relu + segment-max
__global__ void k_edge_alpha(const int* __restrict__ ei,
                             const float* __restrict__ eattr, const float* __restrict__ loop_attr,
                             const float* __restrict__ as_n, const float* __restrict__ ad_n,
                             const float* __restrict__ q,
                             float* __restrict__ alpha, float* __restrict__ mbuf,
                             int E2, int heads) {
  int i = blockIdx.x * blockDim.x + threadIdx.x;
  if (i >= E2 * heads) return;
  int e = i / heads, hd = i - e * heads;
  int sN, dN;
  const float* ea;
  if (e < NEDGE) { sN = ei[e]; dN = ei[NEDGE + e]; ea = eattr + (size_t)e * EDIM; }
  else           { sN = dN = e - NEDGE;            ea = loop_attr + (size_t)(e - NEDGE) * EDIM; }
  float av = 0.0f;
#pragma unroll
  for (int d = 0; d < EDIM; ++d) av += ea[d] * q[d * heads + hd];
  float a = as_n[(size_t)sN * heads + hd] + ad_n[(size_t)dN * heads + hd] + av;
  a = (a > 0.0f) ? a : 0.2f * a;
  alpha[i] = a;
  atomicMaxF(&mbuf[(size_t)dN * heads + hd], a);
}

// exp(alpha - max) + segment-sum of denominators
__global__ void k_edge_exp(const int* __restrict__ ei, const float* __restrict__ mbuf,
                           float* __restrict__ alpha, float* __restrict__ den,
                           int E2, int heads) {
  int i = blockIdx.x * blockDim.x + threadIdx.x;
  if (i >= E2 * heads) return;
  int e = i / heads, hd = i - e * heads;
  int dN = (e < NEDGE) ? ei[NEDGE + e] : (e - NEDGE);
  float ex = __expf(alpha[i] - mbuf[(size_t)dN * heads + hd]);
  alpha[i] = ex;
  atomicAdd(&den[(size_t)dN * heads + hd], ex);
}

// weighted aggregation: out[dst] += (ex/den) * feat[src]
// one block per edge; float4 gathers (rows are 16B aligned), 4 atomics/thread
__global__ void k_aggregate(const int* __restrict__ ei,
                            const float* __restrict__ feat, const float* __restrict__ alpha,
                            const float* __restrict__ den,
                            float* __restrict__ out, int heads, int ch) {
  int e = blockIdx.x;
  int sN, dN;
  if (e < NEDGE) { sN = ei[e]; dN = ei[NEDGE + e]; }
  else           { sN = dN = e - NEDGE; }
  const int hc_total = heads * ch;
  __shared__ float w[8];
  if (threadIdx.x < (unsigned)heads)
    w[threadIdx.x] = alpha[(size_t)e * heads + threadIdx.x] /
                     den[(size_t)dN * heads + threadIdx.x];
  __syncthreads();
  const float4* fs = (const float4*)(feat + (size_t)sN * hc_total);
  float* od = out + (size_t)dN * hc_total;
  for (int qd = threadIdx.x; qd * 4 < hc_total; qd += blockDim.x) {
    float4 v = fs[qd];
    float ww = w[(qd * 4) / ch];      // head boundaries are 4-aligned
    atomicAdd(&od[qd * 4 + 0], ww * v.x);
    atomicAdd(&od[qd * 4 + 1], ww * v.y);
    atomicAdd(&od[qd * 4 + 2], ww * v.z);
    atomicAdd(&od[qd * 4 + 3], ww * v.w);
  }
}

// bias + ELU (layers 1,2), in place
__global__ void k_bias_elu(float* __restrict__ h, const float* __restrict__ b, int hc_total) {
  int i = blockIdx.x * blockDim.x + threadIdx.x;
  if (i >= NNODE * hc_total) return;
  float v = h[i] + b[i % hc_total];
  h[i] = (v > 0.0f) ? v : (__expf(v) - 1.0f);
}

// mean pool: accumulate per-graph sums + counts
__global__ void k_pool(const float* __restrict__ h3, const int* __restrict__ batch,
                       float* __restrict__ pooled, float* __restrict__ gcnt) {
  int n = blockIdx.x;
  int c = threadIdx.x;
  int g = batch[n];
  atomicAdd(&pooled[(size_t)g * COUT + c], h3[(size_t)n * COUT + c]);
  if (c == 0) atomicAdd(&gcnt[g], 1.0f);
}

// divide + fold in b3 (mean of (x+b) = mean(x)+b)
__global__ void k_pool_div(float* __restrict__ pooled, const float* __restrict__ gcnt,
                           const float* __restrict__ b3) {
  int i = blockIdx.x * blockDim.x + threadIdx.x;
  if (i >= NGRAPH * COUT) return;
  int g = i / COUT, c = i - g * COUT;
  float cnt = gcnt[g];
  pooled[i] = (cnt > 0.0f) ? (pooled[i] / cnt + b3[c]) : 0.0f;
}

// ---------------------------------------------------------------------------
extern "C" void kernel_launch(void* const* d_in, const int* in_sizes, int n_in,
                              void* d_out, int out_size, void* d_ws, size_t ws_size,
                              hipStream_t stream) {
  (void)in_sizes; (void)n_in; (void)out_size; (void)ws_size;

  const float* x     = (const float*)d_in[0];
  const int*   ei    = (const int*)d_in[1];
  const float* eattr = (const float*)d_in[2];
  const int*   batch = (const int*)d_in[3];
  const float* W[3]  = {(const float*)d_in[4],  (const float*)d_in[10], (const float*)d_in[16]};
  const float* As[3] = {(const float*)d_in[5],  (const float*)d_in[11], (const float*)d_in[17]};
  const float* Ad[3] = {(const float*)d_in[6],  (const float*)d_in[12], (const float*)d_in[18]};
  const float* We[3] = {(const float*)d_in[7],  (const float*)d_in[13], (const float*)d_in[19]};
  const float* Ae[3] = {(const float*)d_in[8],  (const float*)d_in[14], (const float*)d_in[20]};
  const float* Bb[3] = {(const float*)d_in[9],  (const float*)d_in[15], (const float*)d_in[21]};

  const int E2 = NEDGE + NNODE;

  // workspace carve-out
  char* p = (char*)d_ws;
  auto alloc = [&](size_t nfloats) {
    float* r = (float*)p;
    p += ((nfloats * sizeof(float) + 255) & ~(size_t)255);
    return r;
  };
  float* bufA      = alloc((size_t)NNODE * HC);   // GEMM-transformed features
  float* bufB      = alloc((size_t)NNODE * HC);   // layer input / aggregation output
  float* loop_attr = alloc((size_t)NNODE * EDIM);
  float* deg       = alloc(NNODE);
  float* as_n      = alloc((size_t)NNODE * HEADS);
  float* ad_n      = alloc((size_t)NNODE * HEADS);
  float* mbuf      = alloc((size_t)NNODE * HEADS);
  float* den       = alloc((size_t)NNODE * HEADS);
  float* alpha     = alloc((size_t)E2 * HEADS);
  float* qbuf      = alloc(EDIM * HEADS);
  float* gcnt      = alloc(NGRAPH);

  // ---- self-loop edge attrs (once; reused by all layers) ----
  hipMemsetAsync(loop_attr, 0, (size_t)NNODE * EDIM * sizeof(float), stream);
  hipMemsetAsync(deg, 0, (size_t)NNODE * sizeof(float), stream);
  k_loop_acc<<<(NEDGE + 255) / 256, 256, 0, stream>>>(ei, eattr, loop_attr, deg);
  k_loop_div<<<(NNODE * EDIM + 255) / 256, 256, 0, stream>>>(loop_attr, deg);

  auto run_layer = [&](const float* in, int K, int heads, int ch, int li, bool elu) {
    const int M = heads * ch;
    // dense GEMM: feat = in @ W   (fp32 WMMA, 64x64 strip per wave)
    dim3 gblk(32, 2);
    dim3 ggrd(M / 128, (NNODE + 63) / 64);
    k_gemm_f32wmma<<<ggrd, gblk, 0, stream>>>(in, W[li], bufA, NNODE, K, M);
    // attention scalars
    k_compute_q<<<1, 64, 0, stream>>>(We[li], Ae[li], qbuf, heads, ch);
    k_node_alpha<<<(NNODE * heads + 255) / 256, 256, 0, stream>>>(bufA, As[li], Ad[li],
                                                                  as_n, ad_n, heads, ch);
    // segment softmax over dst
    k_fill<<<(NNODE * heads + 255) / 256, 256, 0, stream>>>(mbuf, -1e30f, NNODE * heads);
    hipMemsetAsync(den, 0, (size_t)NNODE * heads * sizeof(float), stream);
    int eth = E2 * heads;
    k_edge_alpha<<<(eth + 255) / 256, 256, 0, stream>>>(ei, eattr, loop_attr, as_n, ad_n,
                                                        qbuf, alpha, mbuf, E2, heads);
    k_edge_exp<<<(eth + 255) / 256, 256, 0, stream>>>(ei, mbuf, alpha, den, E2, heads);
    // aggregation (bufB already fully consumed by the GEMM above -> safe to zero)
    hipMemsetAsync(bufB, 0, (size_t)NNODE * M * sizeof(float), stream);
    int ablk = M / 4;
    if (ablk > 128) ablk = 128;
    k_aggregate<<<E2, ablk, 0, stream>>>(ei, bufA, alpha, den, bufB, heads, ch);
    if (elu)
      k_bias_elu<<<(NNODE * M + 255) / 256, 256, 0, stream>>>(bufB, Bb[li], M);
  };

  run_layer(x,    CIN, HEADS, CHID, 0, true);   // layer 1: 128 -> 8x64
  run_layer(bufB, HC,  HEADS, CHID, 1, true);   // layer 2: 512 -> 8x64
  run_layer(bufB, HC,  1,     COUT, 2, false);  // layer 3: 512 -> 128 (heads=1)

  // ---- global mean pool (b3 folded into the divide) ----
  hipMemsetAsync(d_out, 0, (size_t)NGRAPH * COUT * sizeof(float), stream);
  hipMemsetAsync(gcnt, 0, (size_t)NGRAPH * sizeof(float), stream);
  k_pool<<<NNODE, COUT, 0, stream>>>(bufB, batch, (float*)d_out, gcnt);
  k_pool_div<<<(NGRAPH * COUT + 255) / 256, 256, 0, stream>>>((float*)d_out, gcnt, Bb[2]);
}